// SGA_89395449299017
// MI455X (gfx1250) — compile-verified
//
#include <hip/hip_runtime.h>
#include <float.h>

// SGA (semi-global aggregation), 4 directions fused into 2 kernels.
// B=1, C=32, D=48, H=96, W=192. Memory-bound (~750MB -> ~32us floor at
// 23.3 TB/s); the recurrence is sequential + nonlinear (max over D) so WMMA
// is inapplicable. CDNA5 paths used:
//   * TDM tensor_load_to_lds (TENSORcnt): one descriptor-driven DMA stages the
//     whole x tile per row, with hardware LDS padding providing the bank-skew.
//   * async global->LDS b32 (ASYNCcnt): stages the g weights transposed,
//     running concurrently with the TDM.
// Each workgroup = 2 waves: wave0 forward scan, wave1 backward scan
// (independent recurrences run concurrently, halving serial latency).

namespace {
constexpr int C = 32, D = 48, H = 96, W = 192;
constexpr int HW = H * W;       // 18432
constexpr int PAIRS = D / 2;    // 24 active lanes per wave, 2 depth values each
// TDM pads +1 DWORD every 64 (k1) / 32 (k2) DWORDs -> skewed row strides:
constexpr int XR1 = W + W / 64; // 195: (2l*195) mod 64 = 6l -> conflict-free columns
constexpr int XR2 = H + H / 32; // 99:  (2l*99)  mod 64 = 6l -> conflict-free columns
constexpr int AP = 50;          // history stride: even (float2-aligned), gcd(50,64)=2 ->
                                // conflict-free for 32 consecutive positions per wave
}

#if defined(__has_builtin)
# if __has_builtin(__builtin_amdgcn_tensor_load_to_lds)
#  define HAVE_TDM 1
# endif
# if __has_builtin(__builtin_amdgcn_s_wait_tensorcnt)
#  define HAVE_TWAIT 1
# endif
#endif

// ---- CDNA5 async global->LDS copy (ASYNCcnt-tracked, no VGPR round trip) ----
__device__ __forceinline__ void async_g2l_b32(void* lds, const void* gaddr) {
  asm volatile("global_load_async_to_lds_b32 %0, %1, off"
               :: "v"((unsigned)(unsigned long long)lds),
                  "v"((unsigned long long)gaddr)
               : "memory");
}
__device__ __forceinline__ void async_wait0() {
  asm volatile("s_wait_asynccnt 0" ::: "memory");
}
__device__ __forceinline__ void tensor_wait0() {
#if defined(HAVE_TWAIT)
  __builtin_amdgcn_s_wait_tensorcnt(0);
#else
  asm volatile("s_wait_tensorcnt 0" ::: "memory");
#endif
}

#if defined(HAVE_TDM)
typedef unsigned u32x4 __attribute__((ext_vector_type(4)));
typedef int i32x8 __attribute__((ext_vector_type(8)));
typedef int i32x4 __attribute__((ext_vector_type(4)));

// Issue one TDM load: tile (tileD0 x tileD1 [x tileD2]) of 4-byte elements,
// dim0 contiguous, dim1 stride s0, dim2 stride s1 (elements), LDS padding of
// +1 DWORD every (1<<padCode)*2 DWORDs. tensor dims = tile dims (no OOB).
__device__ __forceinline__ void tdm_load(void* lds, const void* gaddr,
                                         int tileD0, int tileD1, int tileD2,
                                         long long s0, long long s1, int padCode) {
  unsigned ldsb = (unsigned)__builtin_amdgcn_readfirstlane(
      (int)(unsigned)(unsigned long long)lds);
  unsigned long long ga = (unsigned long long)gaddr;
  u32x4 g0;
  g0[0] = 1u;                                              // count=1, user mode
  g0[1] = ldsb;                                            // lds_addr (bytes)
  g0[2] = (unsigned)ga;                                    // global_addr[31:0]
  g0[3] = (unsigned)((ga >> 32) & 0x1FFFFFFu) | (2u << 30); // addr[56:32] | type=2
  i32x8 g1;
  g1[0] = (2 << 16) | (1 << 20) | (padCode << 22);         // 4B, pad_en, +1 DW pad
  g1[1] = tileD0 << 16;                                    // tensor_dim0[15:0]
  g1[2] = tileD1 << 16;                                    // tensor_dim1[15:0]
  g1[3] = tileD0 << 16;                                    // tile_dim0
  g1[4] = tileD1 | (tileD2 << 16);                         // tile_dim1, tile_dim2
  g1[5] = (int)(s0 & 0xFFFFFFFFll);                        // dim0_stride[31:0]
  g1[6] = (int)((s0 >> 32) & 0xFFFF) | (int)((s1 & 0xFFFF) << 16);
  g1[7] = (int)((s1 >> 16) & 0xFFFFFFFFll);                // dim1_stride[47:16]
  i32x4 g2 = { tileD2, 0, 0, 0 };                          // tensor_dim2 (3D case)
  i32x4 g3 = { 0, 0, 0, 0 };
  i32x8 g4 = { 0, 0, 0, 0, 0, 0, 0, 0 };                   // no chained descriptor
  __builtin_amdgcn_tensor_load_to_lds(g0, g1, g2, g3, g4, 0);
}
#endif

// ---- wave32 max over the 48 depth values (lanes >= PAIRS carry -FLT_MAX) ----
__device__ __forceinline__ float wave_max(float m) {
  m = fmaxf(m, __shfl_xor(m, 16));
  m = fmaxf(m, __shfl_xor(m, 8));
  m = fmaxf(m, __shfl_xor(m, 4));
  m = fmaxf(m, __shfl_xor(m, 2));
  m = fmaxf(m, __shfl_xor(m, 1));
  return m;
}

// One recurrence step. Lane l owns depths d0=2l, d1=2l+1 (l < PAIRS).
// A_new[d] = w0*C[d] + w1*A[d] + w2*A[d-1] + w3*A[d+1] + w4*max_d A
__device__ __forceinline__ void sga_step(int lane, bool act,
                                         float c0, float c1,
                                         const float* gRow,  // [w0..w4]
                                         float& a0, float& a1) {
  float4 wv = *(const float4*)gRow;  // w0,w1,w2,w3 (32B-aligned broadcast read)
  float w4v = gRow[4];
  float up = __shfl_up(a1, 1);       // A[d0-1] from lane l-1
  float dn = __shfl_down(a0, 1);     // A[d1+1] from lane l+1
  float m = wave_max(act ? fmaxf(a0, a1) : -FLT_MAX);
  float dm1_0 = (lane == 0) ? a0 : up;          // edge clamp d=0
  float dp1_1 = (lane == PAIRS - 1) ? a1 : dn;  // edge clamp d=47
  float na0 = wv.x * c0 + wv.y * a0 + wv.z * dm1_0 + wv.w * a1 + w4v * m;
  float na1 = wv.x * c1 + wv.y * a1 + wv.z * a0 + wv.w * dp1_1 + w4v * m;
  a0 = na0; a1 = na1;
}

// One directional scan (dp=+1 from p0=0, or dp=-1 from p0=LEN-1), history->aout.
// xs: skewed [D][XROW] (element (d,p) at d*XROW + p + (p>>SH));
// gs: [LEN][8]; aout: [LEN][AP].
template <int LEN, int XROW, int SH>
__device__ __forceinline__ void sga_scan_dir(const float* xs, const float* gs,
                                             float* aout, int lane, int dp, int p0) {
  const bool act = lane < PAIRS;
  const int d0 = 2 * lane, d1 = d0 + 1;
  int off = p0 + (p0 >> SH);
  float a0 = act ? xs[d0 * XROW + off] : 0.f;   // A(p0) = C(p0)
  float a1 = act ? xs[d1 * XROW + off] : 0.f;
  if (act) *(float2*)&aout[p0 * AP + d0] = make_float2(a0, a1);
  int p = p0;
  for (int t = 1; t < LEN; ++t) {
    p += dp;
    off = p + (p >> SH);
    float c0 = act ? xs[d0 * XROW + off] : 0.f;
    float c1 = act ? xs[d1 * XROW + off] : 0.f;
    sga_step(lane, act, c0, c1, &gs[p * 8], a0, a1);
    if (act) *(float2*)&aout[p * AP + d0] = make_float2(a0, a1);
  }
}

// ---- Kernel 1: W-direction scans (g0 fwd, g1 bwd); out = max(a0,a1) ----
__global__ __launch_bounds__(64)
void sga_w_kernel(const float* __restrict__ x, const float* __restrict__ g0,
                  const float* __restrict__ g1, float* __restrict__ out) {
  __shared__ __align__(16) float xs[D * XR1];   // 37.4 KB (TDM-skewed)
  __shared__ __align__(16) float afwd[W * AP];  // 38.4 KB
  __shared__ __align__(16) float abwd[W * AP];  // 38.4 KB
  __shared__ __align__(16) float gsF[W * 8];    // 6.1 KB
  __shared__ __align__(16) float gsB[W * 8];    // 6.1 KB  -> ~126 KB, 2 WG/WGP
  const int tid = (int)threadIdx.x;             // 0..63
  const int wid = tid >> 5, lane = tid & 31;
  const int row = (int)blockIdx.x;              // c*H + h
  const int cc = row / H, hh = row % H;

  const float* xrow = x + (size_t)cc * D * HW + (size_t)hh * W;
#if defined(HAVE_TDM)
  // One DMA: 192x48 tile, dim1 stride HW, +1 DW LDS pad per 64 DW (code 5)
  if (wid == 0) tdm_load(xs, xrow, W, D, 0, (long long)HW, 0ll, 5);
#else
  for (int d = 0; d < D; ++d) {
    const float* src = xrow + (size_t)d * HW;
    float* dst = &xs[d * XR1];
#pragma unroll
    for (int j = 0; j < W / 64; ++j) {
      int w = tid + 64 * j;
      async_g2l_b32(dst + w + (w >> 6), src + w);
    }
  }
#endif
  // stage g0/g1 transposed to [w][8] via the async engine (overlaps the TDM)
  const float* g0row = g0 + (size_t)cc * 5 * HW + (size_t)hh * W;
  const float* g1row = g1 + (size_t)cc * 5 * HW + (size_t)hh * W;
  for (int k = 0; k < 5; ++k) {
    const float* s0 = g0row + (size_t)k * HW;
    const float* s1 = g1row + (size_t)k * HW;
#pragma unroll
    for (int j = 0; j < W / 64; ++j) {
      int w = tid + 64 * j;
      async_g2l_b32(&gsF[w * 8 + k], s0 + w);
      async_g2l_b32(&gsB[w * 8 + k], s1 + w);
    }
  }
  async_wait0();
#if defined(HAVE_TDM)
  if (wid == 0) tensor_wait0();
#endif
  __syncthreads();

  // wave0: forward scan -> afwd ; wave1: backward scan -> abwd (concurrent)
  if (wid == 0) sga_scan_dir<W, XR1, 6>(xs, gsF, afwd, lane, +1, 0);
  else          sga_scan_dir<W, XR1, 6>(xs, gsB, abwd, lane, -1, W - 1);
  __syncthreads();

  // writeout: max(fwd,bwd), coalesced 128B/wave stores; LDS reads conflict-free
  float* orow = out + (size_t)cc * D * HW + (size_t)hh * W;
  for (int d = 0; d < D; ++d) {
    float* od = orow + (size_t)d * HW;
#pragma unroll
    for (int j = 0; j < W / 64; ++j) {
      int w = tid + 64 * j;
      od[w] = fmaxf(afwd[w * AP + d], abwd[w * AP + d]);
    }
  }
}

// ---- Kernel 2: H-direction scans (g2 fwd, g3 bwd); out = max(out, max(a2,a3)) ----
__global__ __launch_bounds__(64)
void sga_h_kernel(const float* __restrict__ x, const float* __restrict__ g2,
                  const float* __restrict__ g3, float* __restrict__ out) {
  __shared__ __align__(16) float xs[D * XR2];   // 19.0 KB (TDM-skewed)
  __shared__ __align__(16) float afwd[H * AP];  // 19.2 KB
  __shared__ __align__(16) float abwd[H * AP];  // 19.2 KB
  __shared__ __align__(16) float gsF[H * 8];    // 3.1 KB
  __shared__ __align__(16) float gsB[H * 8];    // 3.1 KB -> ~64 KB, 5 WG/WGP
  const int tid = (int)threadIdx.x;
  const int wid = tid >> 5, lane = tid & 31;
  const int row = (int)blockIdx.x;              // c*W + w
  const int cc = row / W, ww = row % W;

  // x gather at fixed w: strides W (h) and HW (d); L2-resident after kernel 1
  const float* xcol = x + (size_t)cc * D * HW + ww;
#if defined(HAVE_TDM)
  // 3D DMA: tile 1 x 96 x 48, strides W and HW, +1 DW pad per 32 DW (code 4)
  if (wid == 0) tdm_load(xs, xcol, 1, H, D, (long long)W, (long long)HW, 4);
#else
  for (int d = 0; d < D; ++d) {
    const float* src = xcol + (size_t)d * HW;
    float* dst = &xs[d * XR2];
    {
      int h = tid;
      async_g2l_b32(dst + h + (h >> 5), src + (size_t)h * W);
    }
    if (tid < 32) {
      int h = 64 + tid;
      async_g2l_b32(dst + h + (h >> 5), src + (size_t)h * W);
    }
  }
#endif
  const float* g2c = g2 + (size_t)cc * 5 * HW + ww;
  const float* g3c = g3 + (size_t)cc * 5 * HW + ww;
  for (int k = 0; k < 5; ++k) {
    const float* s2 = g2c + (size_t)k * HW;
    const float* s3 = g3c + (size_t)k * HW;
    {
      int h = tid;
      async_g2l_b32(&gsF[h * 8 + k], s2 + (size_t)h * W);
      async_g2l_b32(&gsB[h * 8 + k], s3 + (size_t)h * W);
    }
    if (tid < 32) {
      int h = 64 + tid;
      async_g2l_b32(&gsF[h * 8 + k], s2 + (size_t)h * W);
      async_g2l_b32(&gsB[h * 8 + k], s3 + (size_t)h * W);
    }
  }
  async_wait0();
#if defined(HAVE_TDM)
  if (wid == 0) tensor_wait0();
#endif
  __syncthreads();

  if (wid == 0) sga_scan_dir<H, XR2, 5>(xs, gsF, afwd, lane, +1, 0);
  else          sga_scan_dir<H, XR2, 5>(xs, gsB, abwd, lane, -1, H - 1);
  __syncthreads();

  // RMW max with the W-direction result already in d_out (L2-resident)
  float* ocol = out + (size_t)cc * D * HW + ww;
  for (int d = 0; d < D; ++d) {
    float* od = ocol + (size_t)d * HW;
    {
      int h = tid;
      float* p = od + (size_t)h * W;
      *p = fmaxf(*p, fmaxf(afwd[h * AP + d], abwd[h * AP + d]));
    }
    if (tid < 32) {
      int h = 64 + tid;
      float* p = od + (size_t)h * W;
      *p = fmaxf(*p, fmaxf(afwd[h * AP + d], abwd[h * AP + d]));
    }
  }
}

extern "C" void kernel_launch(void* const* d_in, const int* in_sizes, int n_in,
                              void* d_out, int out_size, void* d_ws, size_t ws_size,
                              hipStream_t stream) {
  (void)in_sizes; (void)n_in; (void)out_size; (void)d_ws; (void)ws_size;
  const float* x  = (const float*)d_in[0];
  const float* g0 = (const float*)d_in[1];
  const float* g1 = (const float*)d_in[2];
  const float* g2 = (const float*)d_in[3];
  const float* g3 = (const float*)d_in[4];
  float* out = (float*)d_out;
  sga_w_kernel<<<C * H, 64, 0, stream>>>(x, g0, g1, out);   // writes max(a0,a1)
  sga_h_kernel<<<C * W, 64, 0, stream>>>(x, g2, g3, out);   // maxes in max(a2,a3)
}